// GridSample_13864154432183
// MI455X (gfx1250) — compile-verified
//
#include <hip/hip_runtime.h>

// Grid-sample bilinear, NHWC, zeros padding, align_corners=False,
// with the reference's snap-to-ceil quirk (SNAP_EPS = 1e-5).
//
// Shapes (fixed by setup_inputs):
//   inputs: [8, 256, 256, 64] f32   (NHWC; per-pixel channel vector = 256 B)
//   grid:   [8, 256, 256, 2]  f32
//   out:    [8, 256, 256, 64] f32
//
// Roofline: ~272 MB total traffic (write 134 MB out NT, read 4 MB grid NT,
// input 134 MB stays resident in 192 MB L2 across the 4x gather reuse)
// => ~12 us floor at 23.3 TB/s. FLOPs negligible; WMMA inapplicable (gather).
//
// Mapping (wave32): each wave processes 2 output pixels.
//   lanes 0..15  -> pixel 2*w+0, channels (lane&15)*4 .. +3  (float4)
//   lanes 16..31 -> pixel 2*w+1, channels (lane&15)*4 .. +3  (float4)
// Every gather is a coalesced 256 B half-wave request (global_load_b128).

typedef float v4f __attribute__((ext_vector_type(4)));
typedef float v2f __attribute__((ext_vector_type(2)));

#define GS_N 8
#define GS_H 256
#define GS_W 256
#define GS_C 64
// H*W = 65536 = 2^16 ; H*W*C = 2^22

__global__ __launch_bounds__(256) void GridSample_gfx1250_kernel(
    const float* __restrict__ in,
    const float* __restrict__ grid,
    float* __restrict__ out,
    int nPix)
{
    const int tid  = blockIdx.x * 256 + threadIdx.x;
    const int lane = tid & 31;
    const int pix  = ((tid >> 5) << 1) + (lane >> 4);   // 2 pixels per wave
    const int c4   = lane & 15;                         // float4 slot in C=64
    if (pix >= nPix) return;

    // Prefetch the grid stream ~64 KB ahead of this wave. Locality 3 =>
    // WGP-scope global_prefetch_b8: pulls into all cache levels (incl. WGP$),
    // since this same WGP consumes it shortly.
    __builtin_prefetch(grid + 2 * (pix + 8192), 0, 3);

    // Streamed read-once: non-temporal grid load (TH=NT).
    v2f g = __builtin_nontemporal_load((const v2f*)grid + pix);

    // Unnormalize (align_corners=False) + snap-to-ceil quirk.
    float x = ((g.x + 1.0f) * (float)GS_W - 1.0f) * 0.5f;
    float y = ((g.y + 1.0f) * (float)GS_H - 1.0f) * 0.5f;
    const float cx = ceilf(x);
    x = (cx - x < 1e-5f) ? cx : x;
    const float cy = ceilf(y);
    y = (cy - y < 1e-5f) ? cy : y;

    const float x1f = floorf(x);
    const float y1f = floorf(y);
    const float fx = x - x1f;
    const float fy = y - y1f;
    const int x1 = (int)x1f, y1 = (int)y1f;
    const int x2 = x1 + 1,   y2 = y1 + 1;

    const float wa = (1.0f - fx) * (1.0f - fy);
    const float wb = (1.0f - fx) * fy;
    const float wc = fx * (1.0f - fy);
    const float wd = fx * fy;

    // Base pointer for this batch image + this lane's 4-channel slot.
    const float* basep = in + ((size_t)(pix >> 16) << 22) + (size_t)(c4 << 2);

    auto gather = [&](int xi, int yi) -> v4f {
        const bool valid = ((unsigned)xi < (unsigned)GS_W) &
                           ((unsigned)yi < (unsigned)GS_H);
        const int xc = min(max(xi, 0), GS_W - 1);
        const int yc = min(max(yi, 0), GS_H - 1);
        v4f p = *(const v4f*)(basep + ((size_t)((yc << 8) + xc) << 6));
        if (!valid) { p.x = 0.0f; p.y = 0.0f; p.z = 0.0f; p.w = 0.0f; }
        return p;
    };

    const v4f A  = gather(x1, y1);  // weight wa
    const v4f B  = gather(x1, y2);  // weight wb
    const v4f Cg = gather(x2, y1);  // weight wc
    const v4f D  = gather(x2, y2);  // weight wd

    v4f r;
    r.x = fmaf(wa, A.x, fmaf(wb, B.x, fmaf(wc, Cg.x, wd * D.x)));
    r.y = fmaf(wa, A.y, fmaf(wb, B.y, fmaf(wc, Cg.y, wd * D.y)));
    r.z = fmaf(wa, A.z, fmaf(wb, B.z, fmaf(wc, Cg.z, wd * D.z)));
    r.w = fmaf(wa, A.w, fmaf(wb, B.w, fmaf(wc, Cg.w, wd * D.w)));

    // Streamed write-once: non-temporal store (TH=NT), keep L2 for inputs.
    __builtin_nontemporal_store(r, (v4f*)out + ((size_t)pix << 4) + c4);
}

extern "C" void kernel_launch(void* const* d_in, const int* in_sizes, int n_in,
                              void* d_out, int out_size, void* d_ws, size_t ws_size,
                              hipStream_t stream) {
    const float* in   = (const float*)d_in[0];   // [8,256,256,64] f32
    const float* grid = (const float*)d_in[1];   // [8,256,256,2]  f32
    float* out        = (float*)d_out;           // [8,256,256,64] f32

    const int nPix    = in_sizes[1] / 2;         // N*Ho*Wo = 524288
    const int nWaves  = (nPix + 1) / 2;          // 2 pixels per wave
    const int threads = nWaves * 32;
    const int blocks  = (threads + 255) / 256;

    GridSample_gfx1250_kernel<<<blocks, 256, 0, stream>>>(in, grid, out, nPix);
}